// LuongAttention_15066745275026
// MI455X (gfx1250) — compile-verified
//
#include <hip/hip_runtime.h>

// ---------------------------------------------------------------------------
// Luong attention, MI455X (gfx1250, wave32).
//   dec_h [32,1024] f32, enc_out [32,2048,1024] f32, src_mask [32,2048] bool,
//   Wa [1024,1024] f32  ->  ctx [32,1024], attn [32,2048]
// Bandwidth-bound: enc_out = 256 MB is the only big tensor. Fused online
// softmax => read it exactly ONCE (~11 us at 23.3 TB/s). q-projection is a
// real GEMM and uses V_WMMA_F32_16X16X4_F32.
// ---------------------------------------------------------------------------

typedef __attribute__((ext_vector_type(2))) float v2f;
typedef __attribute__((ext_vector_type(8))) float v8f;

#define B_      32
#define S_      2048
#define H_      1024
#define NEGV    (-1.0e9f)
#define CHUNKS  8                 // S-chunks per batch (one workgroup each)
#define SCHUNK  (S_ / CHUNKS)     // 256 rows per workgroup
#define NWAVES  8                 // 256 threads = 8 wave32
#define SPW     (SCHUNK / NWAVES) // 32 rows per wave

// workspace layout (in floats)
#define WS_Q      0
#define WS_SCORES (WS_Q + B_ * H_)            // masked scores [B,S]
#define WS_PM     (WS_SCORES + B_ * S_)       // per-(b,chunk) running max
#define WS_PL     (WS_PM + B_ * CHUNKS)       // per-(b,chunk) running sum
#define WS_PCTX   (WS_PL + B_ * CHUNKS)       // per-(b,chunk) ctx partial [H]
#define WS_BM     (WS_PCTX + B_ * CHUNKS * H_)// per-batch max
#define WS_BL     (WS_BM + B_)                // per-batch sum

// ---------------------------------------------------------------------------
// Kernel 1: q = dec_h @ Wa^T  via V_WMMA_F32_16X16X4_F32.
// One wave per 16x16 output tile: 2 M-tiles x 64 N-tiles = 128 waves.
// A(16x4) layout: lanes 0-15 hold M=lane with {K0,K1}; lanes 16-31 hold {K2,K3}.
// B(4x16) layout mirrors it with N across lanes.  C: VGPR r = rows r / r+8.
// ---------------------------------------------------------------------------
__global__ __launch_bounds__(32)
void qproj_wmma(const float* __restrict__ dec_h,
                const float* __restrict__ Wa,
                float* __restrict__ q) {
  const int mt   = blockIdx.x & 1;      // 32 / 16 = 2 row tiles
  const int nt   = blockIdx.x >> 1;     // 1024 / 16 = 64 col tiles
  const int lane = threadIdx.x;
  const int m0 = mt * 16, n0 = nt * 16;
  const int r15 = lane & 15;
  const int kl  = (lane >> 4) << 1;     // 0 for lanes 0-15, 2 for lanes 16-31

  const float* ap = dec_h + (m0 + r15) * H_;  // A row (K-major gather)
  const float* bp = Wa    + (n0 + r15) * H_;  // B[k][n] = Wa[n][k]

  v8f c = {};
  #pragma unroll 4
  for (int k0 = 0; k0 < H_; k0 += 4) {
    v2f a, b;
    a.x = ap[k0 + kl];  a.y = ap[k0 + kl + 1];
    b.x = bp[k0 + kl];  b.y = bp[k0 + kl + 1];
    c = __builtin_amdgcn_wmma_f32_16x16x4_f32(
        /*neg_a=*/false, a, /*neg_b=*/false, b,
        /*c_mod=*/(short)0, c, /*reuse_a=*/false, /*reuse_b=*/false);
  }

  const int mhi = (lane >> 4) << 3;     // rows r (lanes 0-15) / r+8 (16-31)
  #pragma unroll
  for (int r = 0; r < 8; ++r)
    q[(m0 + r + mhi) * H_ + n0 + r15] = c[r];
}

// ---------------------------------------------------------------------------
// Kernel 2: fused scores + online softmax + ctx accumulation, single pass
// over enc_out. Grid (CHUNKS, B), 256 threads. Each wave streams 32 rows:
// coalesced float4 loads, shfl_xor dot reduction, wave-uniform rescale.
// Workgroup merges its 8 wave partials through LDS f32 atomics.
// ---------------------------------------------------------------------------
__global__ __launch_bounds__(256)
void attn_pass(const float* __restrict__ enc,
               const unsigned char* __restrict__ mask,
               const float* __restrict__ q,
               float* __restrict__ scores,
               float* __restrict__ pm,
               float* __restrict__ pl,
               float* __restrict__ pctx) {
  __shared__ float lds_ctx[H_];
  __shared__ float lds_m[NWAVES], lds_l[NWAVES];

  const int b     = blockIdx.y;
  const int chunk = blockIdx.x;
  const int tid   = threadIdx.x;
  const int wave  = tid >> 5;
  const int lane  = tid & 31;

  // preload this lane's slice of q[b,:] (8 x float4 = 32 floats)
  float4 qf[8];
  const float4* q4 = reinterpret_cast<const float4*>(q + b * H_);
  #pragma unroll
  for (int j = 0; j < 8; ++j) qf[j] = q4[j * 32 + lane];

  float  m = -3.0e38f, l = 0.0f;
  float4 acc[8];
  #pragma unroll
  for (int j = 0; j < 8; ++j) acc[j] = make_float4(0.f, 0.f, 0.f, 0.f);

  const int sbase = chunk * SCHUNK + wave;
  const unsigned char* mrow = mask + b * S_;
  const size_t encb = (size_t)b * S_ * H_;

  for (int i = 0; i < SPW; ++i) {
    const int s = sbase + i * NWAVES;
    const float4* row = reinterpret_cast<const float4*>(enc + encb + (size_t)s * H_);
    if (i + 1 < SPW)   // global_prefetch_b8 for the next row this wave owns
      __builtin_prefetch(enc + encb + (size_t)(s + NWAVES) * H_ + lane * 16, 0, 0);

    float4 e[8];
    #pragma unroll
    for (int j = 0; j < 8; ++j) e[j] = row[j * 32 + lane];

    float p = 0.f;
    #pragma unroll
    for (int j = 0; j < 8; ++j)
      p += qf[j].x * e[j].x + qf[j].y * e[j].y + qf[j].z * e[j].z + qf[j].w * e[j].w;
    #pragma unroll
    for (int off = 16; off > 0; off >>= 1) p += __shfl_xor(p, off, 32);

    const float sc = mrow[s] ? p : NEGV;   // masked score (finite, like ref)
    if (lane == 0) scores[b * S_ + s] = sc;

    if (sc > m) {                          // wave-uniform rescale (rare)
      const float r = __expf(m - sc);
      l *= r;
      #pragma unroll
      for (int j = 0; j < 8; ++j) {
        acc[j].x *= r; acc[j].y *= r; acc[j].z *= r; acc[j].w *= r;
      }
      m = sc;
    }
    const float w = __expf(sc - m);
    l += w;
    #pragma unroll
    for (int j = 0; j < 8; ++j) {
      acc[j].x += w * e[j].x; acc[j].y += w * e[j].y;
      acc[j].z += w * e[j].z; acc[j].w += w * e[j].w;
    }
  }

  // ---- merge 8 wave partials inside the workgroup ----
  if (lane == 0) { lds_m[wave] = m; lds_l[wave] = l; }
  for (int h = tid; h < H_; h += 256) lds_ctx[h] = 0.f;
  __syncthreads();

  float M = -3.0e38f;
  #pragma unroll
  for (int w2 = 0; w2 < NWAVES; ++w2) M = fmaxf(M, lds_m[w2]);
  float L = 0.f;
  #pragma unroll
  for (int w2 = 0; w2 < NWAVES; ++w2) L += __expf(lds_m[w2] - M) * lds_l[w2];
  const float f = __expf(m - M);          // this wave's rescale into M

  #pragma unroll
  for (int j = 0; j < 8; ++j) {
    const int h = j * 128 + lane * 4;
    atomicAdd(&lds_ctx[h + 0], f * acc[j].x);   // ds_add_f32
    atomicAdd(&lds_ctx[h + 1], f * acc[j].y);
    atomicAdd(&lds_ctx[h + 2], f * acc[j].z);
    atomicAdd(&lds_ctx[h + 3], f * acc[j].w);
  }
  __syncthreads();

  const int pidx = b * CHUNKS + chunk;
  for (int h = tid; h < H_; h += 256) pctx[(size_t)pidx * H_ + h] = lds_ctx[h];
  if (tid == 0) { pm[pidx] = M; pl[pidx] = L; }
}

// ---------------------------------------------------------------------------
// Kernel 3: merge the CHUNKS partials per batch, emit ctx and (m_b, l_b).
// ---------------------------------------------------------------------------
__global__ __launch_bounds__(256)
void combine_k(const float* __restrict__ pm, const float* __restrict__ pl,
               const float* __restrict__ pctx,
               float* __restrict__ ctx_out,
               float* __restrict__ bm, float* __restrict__ bl) {
  const int b = blockIdx.x, tid = threadIdx.x;   // tid -> 4 h values
  float mb = -3.0e38f;
  #pragma unroll
  for (int j = 0; j < CHUNKS; ++j) mb = fmaxf(mb, pm[b * CHUNKS + j]);
  float lb = 0.f;
  #pragma unroll
  for (int j = 0; j < CHUNKS; ++j) lb += __expf(pm[b * CHUNKS + j] - mb) * pl[b * CHUNKS + j];

  float4 c = make_float4(0.f, 0.f, 0.f, 0.f);
  #pragma unroll
  for (int j = 0; j < CHUNKS; ++j) {
    const float f = __expf(pm[b * CHUNKS + j] - mb);
    const float4 v =
        reinterpret_cast<const float4*>(pctx + (size_t)(b * CHUNKS + j) * H_)[tid];
    c.x += f * v.x; c.y += f * v.y; c.z += f * v.z; c.w += f * v.w;
  }
  const float inv = 1.0f / lb;
  reinterpret_cast<float4*>(ctx_out + b * H_)[tid] =
      make_float4(c.x * inv, c.y * inv, c.z * inv, c.w * inv);
  if (tid == 0) { bm[b] = mb; bl[b] = lb; }
}

// ---------------------------------------------------------------------------
// Kernel 4: attn[b,s] = exp(score - m_b) / l_b
// ---------------------------------------------------------------------------
__global__ __launch_bounds__(256)
void attn_out_k(const float* __restrict__ scores,
                const float* __restrict__ bm, const float* __restrict__ bl,
                float* __restrict__ attn) {
  const int idx = blockIdx.x * 256 + threadIdx.x;  // [0, B*S)
  const int b = idx >> 11;                         // S = 2048
  attn[idx] = __expf(scores[idx] - bm[b]) / bl[b];
}

// ---------------------------------------------------------------------------
extern "C" void kernel_launch(void* const* d_in, const int* in_sizes, int n_in,
                              void* d_out, int out_size, void* d_ws, size_t ws_size,
                              hipStream_t stream) {
  const float*         dec_h = (const float*)d_in[0];
  const float*         enc   = (const float*)d_in[1];
  const unsigned char* mask  = (const unsigned char*)d_in[2];  // jnp.bool_
  const float*         Wa    = (const float*)d_in[3];

  float* ws     = (float*)d_ws;
  float* q      = ws + WS_Q;
  float* scores = ws + WS_SCORES;
  float* pm     = ws + WS_PM;
  float* pl     = ws + WS_PL;
  float* pctx   = ws + WS_PCTX;
  float* bm     = ws + WS_BM;
  float* bl     = ws + WS_BL;

  float* ctx_out  = (float*)d_out;            // [B,H]
  float* attn_out = ctx_out + B_ * H_;        // [B,S]

  qproj_wmma<<<dim3(128), dim3(32), 0, stream>>>(dec_h, Wa, q);
  attn_pass<<<dim3(CHUNKS, B_), dim3(256), 0, stream>>>(enc, mask, q, scores,
                                                        pm, pl, pctx);
  combine_k<<<dim3(B_), dim3(256), 0, stream>>>(pm, pl, pctx, ctx_out, bm, bl);
  attn_out_k<<<dim3((B_ * S_) / 256), dim3(256), 0, stream>>>(scores, bm, bl,
                                                              attn_out);
}